// SparseHead1_54631984005778
// MI455X (gfx1250) — compile-verified
//
#include <hip/hip_runtime.h>

typedef float v2f __attribute__((ext_vector_type(2)));
typedef float v8f __attribute__((ext_vector_type(8)));
typedef float f4v __attribute__((ext_vector_type(4)));

#define T_SEQ 4096
#define E_DIM 64
#define KBLK  64
#define PITCH 68   // 64 + 4 pad: float4-aligned rows, conflict-free column reads

__global__ __launch_bounds__(128) void sparse_block_attn(
    const float* __restrict__ kin,
    const float* __restrict__ qin,
    const float* __restrict__ vin,
    float* __restrict__ out)
{
    __shared__ float Ksh[64 * PITCH];
    __shared__ float Qsh[64 * PITCH];   // re-used as S after stage-1 barrier
    __shared__ float Vsh[64 * PITCH];

    const int wg  = blockIdx.x;
    const int b   = wg >> 6;          // / (T/64)
    const int blk = wg & 63;
    const size_t base = ((size_t)b * T_SEQ + (size_t)blk * KBLK) * E_DIM;

    const int tid = threadIdx.x;

    // ---- cooperative coalesced load: 3 x 64x64 f32, float4 per thread x8 ----
    {
        const f4v* gk = (const f4v*)(kin + base);
        const f4v* gq = (const f4v*)(qin + base);
        const f4v* gv = (const f4v*)(vin + base);
        #pragma unroll
        for (int i = 0; i < 8; ++i) {
            int idx = tid + i * 128;        // 0..1023 float4s
            int row = idx >> 4;             // /16
            int c4  = (idx & 15) << 2;
            f4v kvv = gk[idx];
            f4v qvv = gq[idx];
            f4v vvv = gv[idx];
            *(f4v*)&Ksh[row * PITCH + c4] = kvv;
            *(f4v*)&Qsh[row * PITCH + c4] = qvv;
            *(f4v*)&Vsh[row * PITCH + c4] = vvv;
        }
    }
    __syncthreads();

    const int lane = tid & 31;
    const int w    = tid >> 5;        // wave id 0..3 -> output row strip
    const int half = lane >> 4;       // 0/1 (K-pair select per WMMA layout)
    const int lr   = lane & 15;       // M (A) / N (B,C,D) index

    // ---- stage 1: S = K Q^T for tiles n <= w, accumulated in registers ----
    v8f sacc[4];
    #pragma unroll
    for (int n = 0; n < 4; ++n) {
        sacc[n] = (v8f){};
        if (n <= w) {                 // wave-uniform: EXEC stays all-ones
            #pragma unroll
            for (int e0 = 0; e0 < 64; e0 += 4) {
                v2f a, bb;
                const float* ap = &Ksh[(w * 16 + lr) * PITCH + e0 + 2 * half];
                a.x = ap[0]; a.y = ap[1];                 // A: K[M, e], M=lr
                const float* bp = &Qsh[(n * 16 + lr) * PITCH + e0 + 2 * half];
                bb.x = bp[0]; bb.y = bp[1];               // B: Q^T[e, N], N=lr
                sacc[n] = __builtin_amdgcn_wmma_f32_16x16x4_f32(
                    false, a, false, bb, (short)0, sacc[n], false, false);
            }
        }
    }

    __syncthreads();                  // everyone done reading Qsh -> alias as S

    float* Sstrip = &Qsh[(w * 16) * PITCH];
    #pragma unroll
    for (int n = 0; n < 4; ++n) {
        if (n <= w) {
            #pragma unroll
            for (int r = 0; r < 8; ++r) {
                int mloc = r + 8 * half;                  // local row in strip
                float val = sacc[n][r];
                if (n == w && lr > mloc) val = 0.0f;      // causal mask on diag tile
                Sstrip[mloc * PITCH + n * 16 + lr] = val;
            }
        }
    }
    // same-wave LDS store->load is in-order; no extra barrier needed.

    // ---- stage 2: Out strip = S(16 x kmax) @ V(kmax x 64) ----
    const int kmax = (w + 1) * 16;    // cols beyond this are masked zeros
    float* op = out + base;
    #pragma unroll
    for (int n = 0; n < 4; ++n) {
        v8f acc = (v8f){};
        #pragma unroll 4
        for (int c0 = 0; c0 < kmax; c0 += 4) {
            v2f a, bb;
            const float* ap = &Sstrip[lr * PITCH + c0 + 2 * half];
            a.x = ap[0]; a.y = ap[1];                     // A: S[M, c]
            const float* bp = &Vsh[(c0 + 2 * half) * PITCH + n * 16 + lr];
            bb.x = bp[0]; bb.y = bp[PITCH];               // B: V[c, N] rows c, c+1
            acc = __builtin_amdgcn_wmma_f32_16x16x4_f32(
                false, a, false, bb, (short)0, acc, false, false);
        }
        #pragma unroll
        for (int r = 0; r < 8; ++r) {
            int mloc = r + 8 * half;
            op[(w * 16 + mloc) * E_DIM + n * 16 + lr] = acc[r];
        }
    }
}

extern "C" void kernel_launch(void* const* d_in, const int* in_sizes, int n_in,
                              void* d_out, int out_size, void* d_ws, size_t ws_size,
                              hipStream_t stream) {
    const float* k = (const float*)d_in[0];
    const float* q = (const float*)d_in[1];
    const float* v = (const float*)d_in[2];
    float* out = (float*)d_out;
    dim3 grid(32 * 64);   // B * (T / K_BLOCK) = 2048 independent block tasks
    dim3 block(128);      // 4 waves: one 16-row output strip each
    hipLaunchKernelGGL(sparse_block_attn, grid, block, 0, stream, k, q, v, out);
}